// VoltageNet_67164698575377
// MI455X (gfx1250) — compile-verified
//
#include <hip/hip_runtime.h>
#include <hip/hip_bf16.h>
#include <math.h>

typedef float v2f __attribute__((ext_vector_type(2)));
typedef float v8f __attribute__((ext_vector_type(8)));

#define THREADS 256
#define WAVES 8

__constant__ float c_LB[8] = {0.0f, 0.0f, 0.05f, 0.0f, 0.0f, 0.04236f, 0.023f, 0.0f};
__constant__ float c_UB[8] = {0.04f, 1e-06f, 0.8f, 0.0f, 0.0f, 0.84f, 0.84804f, 0.0f};

// Branchless, stable softplus: max(z,0) + log(1 + exp(-|z|))
__device__ __forceinline__ float softplus_f(float z) {
    return fmaxf(z, 0.0f) + __logf(1.0f + __expf(-fabsf(z)));
}
__device__ __forceinline__ float sigmoid_f(float z) {
    return 1.0f / (1.0f + __expf(-z));
}
// Clamped fast tanh (arguments here reach |x|~60; tanh saturates by 15)
__device__ __forceinline__ float tanh_f(float x) {
    float xc = fminf(fmaxf(x, -15.0f), 15.0f);
    float e2 = __expf(2.0f * xc);
    return (e2 - 1.0f) / (e2 + 1.0f);
}

// ---------------------------------------------------------------------------
// K1: per-battery nanmean(Temp) and prefix-sum SOC
// ---------------------------------------------------------------------------
__global__ __launch_bounds__(THREADS)
void k1_soc_tmean(const float* __restrict__ X, const float* __restrict__ SC,
                  float* __restrict__ soc, float* __restrict__ tmean,
                  int B, int T)
{
    const int b   = blockIdx.x;
    const int tid = threadIdx.x;
    const float* Xb = X + (size_t)b * T * 3;

    __shared__ float s_a[THREADS];
    __shared__ float s_b[THREADS];

    // ---- nanmean of Temp (feature 2) ----
    float sum = 0.0f, cnt = 0.0f;
    for (int t = tid; t < T; t += THREADS) {
        float v = Xb[t * 3 + 2];
        if (!isnan(v)) { sum += v; cnt += 1.0f; }
    }
    s_a[tid] = sum; s_b[tid] = cnt;
    __syncthreads();
    for (int off = THREADS / 2; off > 0; off >>= 1) {
        if (tid < off) { s_a[tid] += s_a[tid + off]; s_b[tid] += s_b[tid + off]; }
        __syncthreads();
    }
    if (tid == 0) tmean[b] = s_a[0] / s_b[0];
    __syncthreads();

    // ---- SOC = Q/QN + cumsum(dSOC), dSOC[t] = (I[t]+I[t-1])(tt[t]-tt[t-1])/(2*3600*QN)
    const float kinv = 1.0f / (2.0f * 3600.0f * 5.0f);
    const int chunk = (T + THREADS - 1) / THREADS;
    const int t0 = tid * chunk;
    float local = 0.0f;
    for (int j = 0; j < chunk; ++j) {
        int t = t0 + j;
        if (t >= 1 && t < T) {
            float ds = (Xb[t*3+1] + Xb[(t-1)*3+1]) * (Xb[t*3+0] - Xb[(t-1)*3+0]) * kinv;
            local += ds;
        }
    }
    s_a[tid] = local;
    __syncthreads();
    // inclusive Hillis–Steele scan of per-thread partials
    for (int off = 1; off < THREADS; off <<= 1) {
        float add = (tid >= off) ? s_a[tid - off] : 0.0f;
        __syncthreads();
        s_a[tid] += add;
        __syncthreads();
    }
    float run = SC[b*2+0] * (1.0f / 5.0f) + ((tid == 0) ? 0.0f : s_a[tid - 1]);
    for (int j = 0; j < chunk; ++j) {
        int t = t0 + j;
        if (t < T) {
            if (t >= 1) {
                run += (Xb[t*3+1] + Xb[(t-1)*3+1]) * (Xb[t*3+0] - Xb[(t-1)*3+0]) * kinv;
            }
            soc[(size_t)b * T + t] = run;
        }
    }
}

// ---------------------------------------------------------------------------
// K2: WMMA MLP (3->6 softplus ->7), theta, OCV curve, scan coefficients.
// One wave processes a 16-timestep tile with V_WMMA_F32_16X16X4_F32.
// ---------------------------------------------------------------------------
__global__ __launch_bounds__(THREADS)
void k2_mlp_wmma(const float* __restrict__ X, const float* __restrict__ SC,
                 const float* __restrict__ soc, const float* __restrict__ tmean,
                 const float* __restrict__ W1, const float* __restrict__ b1,
                 const float* __restrict__ W2, const float* __restrict__ b2,
                 float4* __restrict__ coeff, float* __restrict__ ocv,
                 float* __restrict__ ocvu0, int B, int T)
{
    __shared__ float lds[WAVES][16][17];   // per-wave 16x16 tile (padded)

    const int b    = blockIdx.x;
    const int lane = threadIdx.x & 31;
    const int wave = threadIdx.x >> 5;
    const int col  = lane & 15;
    const bool hi  = lane >= 16;

    const float* Xb = X + (size_t)b * T * 3;
    const float R0  = SC[b*2+1];
    const float Tm  = tmean[b];

    // ---- weight matrices in WMMA B layout (K rows x 16 cols, K padded to 4/8) ----
    v2f B1; B1[0] = 0.0f; B1[1] = 0.0f;          // W1 padded to 4x16
    if (col < 6) {
        B1[0] = hi ? W1[2*6 + col] : W1[0*6 + col];   // K=2 / K=0
        B1[1] = hi ? 0.0f          : W1[1*6 + col];   // K=3 / K=1
    }
    v2f B2a; B2a[0] = 0.0f; B2a[1] = 0.0f;       // W2 rows 0..3
    v2f B2b; B2b[0] = 0.0f; B2b[1] = 0.0f;       // W2 rows 4..7 (6,7 = zero pad)
    if (col < 7) {
        B2a[0] = hi ? W2[2*7 + col] : W2[0*7 + col];
        B2a[1] = hi ? W2[3*7 + col] : W2[1*7 + col];
        B2b[0] = hi ? 0.0f          : W2[4*7 + col];
        B2b[1] = hi ? 0.0f          : W2[5*7 + col];
    }
    const float c1v = (col < 6) ? b1[col] : 0.0f;
    const float c2v = (col < 7) ? b2[col] : 0.0f;
    v8f c1init, c2init;
#pragma unroll
    for (int i = 0; i < 8; ++i) { c1init[i] = c1v; c2init[i] = c2v; }

    const float lb = c_LB[col < 7 ? col : 7];
    const float ub = c_UB[col < 7 ? col : 7];

    const int ntiles = T / 16;
    for (int tile = wave; tile < ntiles; tile += WAVES) {
        const int tbase = tile * 16;
        const int m = col;                 // timestep within tile for this lane
        const int tA = tbase + m;

        // A (16x4 f32): K0=SOC, K1=R0, K2=Tmean, K3=0
        v2f A1;
        if (!hi) { A1[0] = soc[(size_t)b * T + tA]; A1[1] = R0; }
        else     { A1[0] = Tm;                      A1[1] = 0.0f; }

        v8f h = __builtin_amdgcn_wmma_f32_16x16x4_f32(
                    false, A1, false, B1, (short)0, c1init, false, false);

        // softplus + transpose through LDS into A layout for layer 2
#pragma unroll
        for (int i = 0; i < 8; ++i)
            lds[wave][i + (hi ? 8 : 0)][col] = softplus_f(h[i]);

        v2f A2a, A2b;
        if (!hi) { A2a[0] = lds[wave][m][0]; A2a[1] = lds[wave][m][1];
                   A2b[0] = lds[wave][m][4]; A2b[1] = lds[wave][m][5]; }
        else     { A2a[0] = lds[wave][m][2]; A2a[1] = lds[wave][m][3];
                   A2b[0] = lds[wave][m][6]; A2b[1] = lds[wave][m][7]; }

        v8f p = __builtin_amdgcn_wmma_f32_16x16x4_f32(
                    false, A2a, false, B2a, (short)0, c2init, false, false);
        p     = __builtin_amdgcn_wmma_f32_16x16x4_f32(
                    false, A2b, false, B2b, (short)0, p,      false, false);

        // theta = LB + (UB-LB)*sigmoid(0.01*p), store transposed tile
#pragma unroll
        for (int i = 0; i < 8; ++i) {
            float s = sigmoid_f(0.01f * p[i]);
            lds[wave][i + (hi ? 8 : 0)][col] = lb + (ub - lb) * s;
        }

        // per-timestep phase: lane's (col) indexes the timestep now
        const int t  = tbase + m;
        const float R1   = lds[wave][m][0];
        const float Cc   = lds[wave][m][1];
        const float OCVU = lds[wave][m][2];
        const float Mh   = lds[wave][m][3];
        const float Kh   = lds[wave][m][4];
        const float xo   = lds[wave][m][5];
        const float yo   = lds[wave][m][6];

        float Up = 4.4167f + yo*(-1.6518f + yo*(1.6225f + yo*(-2.0843f
                 + yo*(3.5146f + yo*(-2.2166f)))))
                 - 4.0f * __expf(109.451f * yo - 100.006f);
        float Un = 0.063f + 0.8f * __expf(-75.0f * (xo + 0.001f))
                 - 0.012f  * tanh_f((xo - 0.127f) * (1.0f/0.016f))
                 - 0.0118f * tanh_f((xo - 0.155f) * (1.0f/0.016f))
                 - 0.0035f * tanh_f((xo - 0.22f)  * (1.0f/0.02f))
                 - 0.0095f * tanh_f((xo - 0.19f)  * (1.0f/0.013f))
                 - 0.0145f * tanh_f((xo - 0.49f)  * (1.0f/0.02f))
                 - 0.08f   * tanh_f((xo - 1.03f)  * (1.0f/0.055f));
        ocv[(size_t)b * T + t] = Up - Un;

        const float It  = Xb[t*3 + 1];
        const float It1 = (t + 1 < T) ? Xb[(t+1)*3 + 1] : It;   // last coeff -> identity
        const float dtc = It1 - It;                              // reference: dt = diff(I)

        float4 cf;
        cf.x = 1.0f + dtc * Kh * It;      // aUH
        cf.y = dtc * Kh * It * Mh;        // bUH
        cf.z = 1.0f - dtc * Cc;           // aU1
        cf.w = dtc * Cc * R1 * It;        // bU1
        coeff[(size_t)b * T + t] = cf;

        if (t == 0) ocvu0[b] = OCVU;
    }
}

// ---------------------------------------------------------------------------
// K3: parallel linear-recurrence scan (compose (a,b) pairs) + output
// ---------------------------------------------------------------------------
__global__ __launch_bounds__(THREADS)
void k3_scan(const float* __restrict__ X, const float* __restrict__ SC,
             const float4* __restrict__ coeff, const float* __restrict__ ocv,
             const float* __restrict__ ocvu0, float* __restrict__ out,
             int B, int T)
{
    __shared__ float4 s_tr[THREADS];

    const int b   = blockIdx.x;
    const int tid = threadIdx.x;
    const float* Xb = X + (size_t)b * T * 3;
    const float R0  = SC[b*2+1];

    const int chunk = (T + THREADS - 1) / THREADS;
    const int s = tid * chunk;

    // local compose of coeffs [s, s+chunk)
    float aU = 1.0f, bU = 0.0f, aV = 1.0f, bV = 0.0f;
    for (int j = 0; j < chunk; ++j) {
        int e = s + j;
        if (e < T) {
            if (j + 8 < chunk) __builtin_prefetch(&coeff[(size_t)b * T + e + 8], 0, 0);
            float4 c = coeff[(size_t)b * T + e];
            bU = c.x * bU + c.y;  aU = c.x * aU;
            bV = c.z * bV + c.w;  aV = c.z * aV;
        }
    }
    s_tr[tid] = make_float4(aU, bU, aV, bV);
    __syncthreads();

    // inclusive compose-scan over thread aggregates
    for (int off = 1; off < THREADS; off <<= 1) {
        float4 cur  = s_tr[tid];
        float4 prev = (tid >= off) ? s_tr[tid - off] : make_float4(1.f, 0.f, 1.f, 0.f);
        __syncthreads();
        float4 comb;
        comb.x = cur.x * prev.x;  comb.y = cur.x * prev.y + cur.y;
        comb.z = cur.z * prev.z;  comb.w = cur.z * prev.w + cur.w;
        s_tr[tid] = comb;
        __syncthreads();
    }
    float4 E = (tid == 0) ? make_float4(1.f, 0.f, 1.f, 0.f) : s_tr[tid - 1]; // exclusive

    // initial state y0, advanced to this thread's segment start
    const float I0  = Xb[1];
    const float U10 = -ocvu0[b] - I0 * R0;
    float UH = E.y;                 // UH0 == 0
    float U1 = E.z * U10 + E.w;

    for (int j = 0; j < chunk; ++j) {
        int t = s + j;
        if (t < T) {
            float It = Xb[t*3 + 1];
            out[(size_t)b * T + t] = ocv[(size_t)b * T + t] + UH + U1 + It * R0;
            float4 c = coeff[(size_t)b * T + t];
            UH = c.x * UH + c.y;
            U1 = c.z * U1 + c.w;
        }
    }
}

// ---------------------------------------------------------------------------
extern "C" void kernel_launch(void* const* d_in, const int* in_sizes, int n_in,
                              void* d_out, int out_size, void* d_ws, size_t ws_size,
                              hipStream_t stream)
{
    const float* X  = (const float*)d_in[0];
    const float* SC = (const float*)d_in[1];
    const float* W1 = (const float*)d_in[2];
    const float* b1 = (const float*)d_in[3];
    const float* W2 = (const float*)d_in[4];
    const float* b2 = (const float*)d_in[5];
    float* out = (float*)d_out;

    const int B = in_sizes[1] / 2;              // SC is (B,2)
    const int T = in_sizes[0] / (3 * B);        // X is (B,T,3)

    // workspace layout (float4 first for alignment)
    float4* coeff = (float4*)d_ws;              // B*T float4
    float*  soc   = (float*)(coeff + (size_t)B * T);
    float*  ocv   = soc   + (size_t)B * T;
    float*  tmean = ocv   + (size_t)B * T;
    float*  ocvu0 = tmean + B;

    k1_soc_tmean<<<B, THREADS, 0, stream>>>(X, SC, soc, tmean, B, T);
    k2_mlp_wmma <<<B, THREADS, 0, stream>>>(X, SC, soc, tmean, W1, b1, W2, b2,
                                            coeff, ocv, ocvu0, B, T);
    k3_scan     <<<B, THREADS, 0, stream>>>(X, SC, coeff, ocv, ocvu0, out, B, T);
}